// LlamaFlashAttention2_AttnPred_33921651703983
// MI455X (gfx1250) — compile-verified
//
#include <hip/hip_runtime.h>
#include <math.h>

// ---------------- problem constants (match reference) ----------------
#define B_        4
#define H_        32
#define KVH_      8
#define GRP_      4
#define D_        128
#define HID_      4096
#define S_CACHE_  8191
#define S_        8192
#define BLKSZ_    16
#define NBLK_     512      // S_/BLKSZ_
#define TOPKB_    64       // TOPK/BLK
#define HIST_     64
#define TW        8        // cnn width tile == waves per workgroup

typedef float v2f __attribute__((ext_vector_type(2)));
typedef float v8f __attribute__((ext_vector_type(8)));

// =====================================================================
// 1) QKV projection, split-K partials (deterministic, no float atomics)
// =====================================================================
__global__ __launch_bounds__(128)
void qkv_partial_kernel(const float* __restrict__ hidden,
                        const float* __restrict__ Wq,
                        const float* __restrict__ Wk,
                        const float* __restrict__ Wv,
                        float* __restrict__ partial)
{
    __shared__ float hid[B_][128];
    const int g = blockIdx.x, hz = blockIdx.y, tid = threadIdx.x;
    const float* W; int outdim, col;
    if (g < 32)      { W = Wq; outdim = HID_; col = g * 128 + tid; }
    else if (g < 40) { W = Wk; outdim = 1024; col = (g - 32) * 128 + tid; }
    else             { W = Wv; outdim = 1024; col = (g - 40) * 128 + tid; }

    float acc0 = 0.f, acc1 = 0.f, acc2 = 0.f, acc3 = 0.f;
    const int hbeg = hz * 512;
    for (int h0 = hbeg; h0 < hbeg + 512; h0 += 128) {
        __syncthreads();
        for (int b = 0; b < B_; ++b) hid[b][tid] = hidden[b * HID_ + h0 + tid];
        __syncthreads();
        __builtin_prefetch(&W[(size_t)(h0 + 128) * outdim + col], 0, 1);
        #pragma unroll 4
        for (int hh = 0; hh < 128; ++hh) {
            float w = W[(size_t)(h0 + hh) * outdim + col];
            acc0 += hid[0][hh] * w;
            acc1 += hid[1][hh] * w;
            acc2 += hid[2][hh] * w;
            acc3 += hid[3][hh] * w;
        }
    }
    const int cg = g * 128 + tid;   // 0..6143 global column
    partial[((size_t)hz * B_ + 0) * 6144 + cg] = acc0;
    partial[((size_t)hz * B_ + 1) * 6144 + cg] = acc1;
    partial[((size_t)hz * B_ + 2) * 6144 + cg] = acc2;
    partial[((size_t)hz * B_ + 3) * 6144 + cg] = acc3;
}

// =====================================================================
// 2) combine split-K partials + RoPE. One workgroup per 128-dim head.
// =====================================================================
__global__ __launch_bounds__(128)
void rope_combine_kernel(const float* __restrict__ partial,
                         const float* __restrict__ cosv,
                         const float* __restrict__ sinv,
                         float* __restrict__ q_rot,
                         float* __restrict__ k_new,
                         float* __restrict__ v_new)
{
    __shared__ float rot[B_][128];
    const int g = blockIdx.x, tid = threadIdx.x;
    const int cg = g * 128 + tid;
    float acc[B_];
    for (int b = 0; b < B_; ++b) {
        float s = 0.f;
        #pragma unroll
        for (int hz = 0; hz < 8; ++hz)
            s += partial[((size_t)hz * B_ + b) * 6144 + cg];
        acc[b] = s;
    }
    if (g >= 40) {                       // V: no RoPE (uniform per block)
        const int kvh = g - 40;
        for (int b = 0; b < B_; ++b) v_new[(b * KVH_ + kvh) * D_ + tid] = acc[b];
        return;
    }
    for (int b = 0; b < B_; ++b) rot[b][tid] = acc[b];
    __syncthreads();
    const int d = tid;
    const int p = (d < 64) ? d + 64 : d - 64;
    const float sgn = (d < 64) ? -1.f : 1.f;
    for (int b = 0; b < B_; ++b) {
        float r = acc[b] * cosv[b * D_ + d] + sgn * rot[b][p] * sinv[b * D_ + d];
        if (g < 32) q_rot[(b * H_ + g) * D_ + d] = r;
        else        k_new[(b * KVH_ + (g - 32)) * D_ + d] = r;
    }
}

// =====================================================================
// 3) fused CNN predictor with WMMA f32 16x16x4 conv2.
//    K permuted so lanes 16-31 read at constant LDS displacement from
//    lanes 0-15 -> B operands are dual-imm-offset ds_load_2addr_b32;
//    B loads software-pipelined 2 deep to hide LDS latency behind WMMA.
// =====================================================================
__global__ __launch_bounds__(256)
void cnn_tsp_kernel(const float* __restrict__ hist,   // (128,64,512)
                    const float* __restrict__ c1w, const float* __restrict__ c1b,
                    const float* __restrict__ c2w, const float* __restrict__ c2b,
                    const float* __restrict__ c3w, const float* __restrict__ c3b,
                    float* __restrict__ tsp)          // (128,512)
{
    __shared__ float xt[66][12];                       // input tile, padded
    __shared__ float o1[16 * 66 * 10];                 // conv1+relu tile, flat
    __shared__ __align__(16) float w2p[32 * 144];      // conv2 weights, K-permuted
    __shared__ float w1s[16 * 9];
    __shared__ float b1s[16], b2s[32], c3s[32];

    const int n  = blockIdx.y;             // sample 0..127 (= b*32+h)
    const int w0 = blockIdx.x * TW;        // first output column of tile
    const int tid = threadIdx.x;

    // stage conv2 weights pre-permuted to the WMMA K order:
    //  kk<32:  c=kk>>2, tp=kk&3 : K(j) = (c1=2c+(j>>1), tap=2tp+(j&1))
    //  kk>=32: c=kk-32          : K(j) = (c1=4c+j,      tap=8)
    for (int i = tid; i < 32 * 144; i += 256) {
        const int m = i / 144, Kp = i % 144;
        const int kk = Kp >> 2, j = Kp & 3;
        int c1, t;
        if (kk < 32) { const int c = kk >> 2, tp = kk & 3;
                       c1 = 2 * c + (j >> 1); t = 2 * tp + (j & 1); }
        else         { c1 = 4 * (kk - 32) + j; t = 8; }
        w2p[i] = c2w[m * 144 + c1 * 9 + t];
    }
    for (int i = tid; i < 144; i += 256) w1s[i] = c1w[i];
    if (tid < 16) b1s[tid] = c1b[tid];
    if (tid < 32) { b2s[tid] = c2b[tid]; c3s[tid] = c3w[tid]; }

    // stage input (rows -1..64 pad, cols w0-2..w0+9): async global->LDS for
    // in-range elements, plain DS stores for the zero padding (disjoint).
    for (int i = tid; i < 66 * 12; i += 256) {
        const int hp = i / 12, wq = i % 12;
        const int hr = hp - 1, wc = w0 - 2 + wq;
        if (hr >= 0 && hr < HIST_ && wc >= 0 && wc < NBLK_) {
            const unsigned lds_addr =
                (unsigned)(unsigned long long)&xt[hp][wq];
            const unsigned long long gaddr = (unsigned long long)
                &hist[((size_t)n * HIST_ + hr) * NBLK_ + wc];
            asm volatile("global_load_async_to_lds_b32 %0, %1, off"
                         :: "v"(lds_addr), "v"(gaddr) : "memory");
        } else {
            xt[hp][wq] = 0.f;
        }
    }
    asm volatile("s_wait_asynccnt 0x0" ::: "memory");
    __syncthreads();

    // conv1 + relu into o1 (16 ch x 66 padded rows x 10 cols with w-halo)
    for (int i = tid; i < 16 * 66 * 10; i += 256) {
        const int c1 = i / 660, rem = i % 660, hp = rem / 10, wp = rem % 10;
        const int wc = w0 - 1 + wp;
        float v = 0.f;
        if (hp >= 1 && hp <= 64 && wc >= 0 && wc < NBLK_) {
            float s = b1s[c1];
            #pragma unroll
            for (int dh = 0; dh < 3; ++dh)
                #pragma unroll
                for (int dw = 0; dw < 3; ++dw)
                    s += w1s[c1 * 9 + dh * 3 + dw] * xt[hp - 1 + dh][wp + dw];
            v = fmaxf(s, 0.f);
        }
        o1[i] = v;
    }
    __syncthreads();

    // conv2: M = c2 half (16), N = 16 h-positions, K = 144 in 36 chunks.
    const int wave = tid >> 5, lane = tid & 31;
    const int wi   = wave;                 // column within tile (0..7)
    const int nloc = lane & 15;            // = M row for A, = N col for B
    const bool lo  = (lane < 16);          // half-wave K split (K0/K1 vs K2/K3)
    float colsum = 0.f;

    for (int Mb = 0; Mb < 2; ++Mb) {
        const v2f* ap = (const v2f*)&w2p[(Mb * 16 + nloc) * 144 + (lo ? 0 : 2)];
        for (int ht = 0; ht < 4; ++ht) {
            const float* bp  = &o1[(ht * 16 + nloc) * 10 + wi];
            const float* bP  = bp + (lo ? 0 : 660);    // chunk type P: Δc1 = 1
            const float* bQ  = bp + (lo ? 0 : 1320);   // chunk type Q: Δc1 = 2
            // B fetch for chunk kk; with constant kk everything folds so the
            // base (bP + 2c*660) is CSE'd across the 4 chunks per channel
            // pair and tap offsets (<=22 elems) fit dual 8-bit imm offsets.
            auto loadB = [&](int kk) -> v2f {
                v2f r;
                if (kk < 32) {
                    const int c = kk >> 2, tp = kk & 3;
                    const int t0 = 2 * tp, t1 = t0 + 1;
                    const float* base = bP + 2 * c * 660;
                    r.x = base[(t0 / 3) * 10 + (t0 % 3)];
                    r.y = base[(t1 / 3) * 10 + (t1 % 3)];
                } else {
                    const int c = kk - 32;             // tap 8 (dh=2,dw=2) -> +22
                    r.x = bQ[(4 * c)     * 660 + 22];
                    r.y = bQ[(4 * c + 1) * 660 + 22];
                }
                return r;
            };
            v8f acc = {0.f, 0.f, 0.f, 0.f, 0.f, 0.f, 0.f, 0.f};
            v2f bcur = loadB(0);
            v2f bnxt = loadB(1);
            #pragma unroll
            for (int kk = 0; kk < 36; ++kk) {
                const v2f a = ap[kk * 2];              // ds_load_b64, imm offset
                const v2f bfut = (kk + 2 < 36) ? loadB(kk + 2) : bcur;
                acc = __builtin_amdgcn_wmma_f32_16x16x4_f32(
                    /*neg_a=*/false, a, /*neg_b=*/false, bcur,
                    /*c_mod=*/(short)0, acc, /*reuse_a=*/false, /*reuse_b=*/false);
                bcur = bnxt;
                bnxt = bfut;
            }
            // D layout: vgpr r -> M = r (lanes 0-15) or r+8 (lanes 16-31)
            const int mrow = lo ? 0 : 8;
            #pragma unroll
            for (int r = 0; r < 8; ++r) {
                const int c2 = Mb * 16 + r + mrow;
                colsum += c3s[c2] * fmaxf(acc[r] + b2s[c2], 0.f);
            }
        }
    }
    // reduce over the 32 lanes (covers all c2 and all 64 h)
    for (int off = 16; off > 0; off >>= 1)
        colsum += __shfl_xor(colsum, off, 32);
    if (lane == 0)
        tsp[(size_t)n * NBLK_ + (w0 + wi)] = colsum * (1.f / 64.f) + c3b[0];
}

// =====================================================================
// 4) exact top-64-of-512 block selection + sink/local, one wave per (b,h)
// =====================================================================
__global__ __launch_bounds__(32)
void topk_mask_kernel(const float* __restrict__ tsp, int* __restrict__ mask)
{
    const int bh = blockIdx.x;      // 0..127
    const int lane = threadIdx.x;   // 0..31
    const float* t = tsp + (size_t)bh * NBLK_;
    float v[16]; int sel[16];
    #pragma unroll
    for (int j = 0; j < 16; ++j) { v[j] = t[lane * 16 + j]; sel[j] = 0; }

    for (int it = 0; it < TOPKB_; ++it) {
        float bv = -INFINITY; int bi = 0x7fffffff;
        #pragma unroll
        for (int j = 0; j < 16; ++j)
            if (v[j] > bv) { bv = v[j]; bi = lane * 16 + j; }
        for (int off = 16; off > 0; off >>= 1) {
            float ov = __shfl_xor(bv, off, 32);
            int   oi = __shfl_xor(bi, off, 32);
            if (ov > bv || (ov == bv && oi < bi)) { bv = ov; bi = oi; }
        }
        const int loc = bi - lane * 16;
        if (loc >= 0 && loc < 16) { sel[loc] = 1; v[loc] = -INFINITY; }
    }
    #pragma unroll
    for (int j = 0; j < 16; ++j) {
        const int blk = lane * 16 + j;
        mask[(size_t)bh * NBLK_ + blk] =
            sel[j] | (blk < 4) | (blk >= NBLK_ - 4);  // SINK=64, LOCAL=64
    }
}

// =====================================================================
// 5) block-sparse decode attention (masked-out probs underflow to 0 ->
//    exact). 8 waves x 2 tokens/block, online softmax, flash-style merge.
// =====================================================================
__global__ __launch_bounds__(256)
void sparse_attn_kernel(const float* __restrict__ k_cache,
                        const float* __restrict__ v_cache,
                        const float* __restrict__ q_rot,
                        const float* __restrict__ k_new,
                        const float* __restrict__ v_new,
                        const int* __restrict__ mask,
                        float* __restrict__ attnout)
{
    const int bh = blockIdx.x;
    const int b = bh >> 5, h = bh & 31, kvh = h >> 2;
    const int tid = threadIdx.x, wave = tid >> 5, lane = tid & 31;
    const float scale = 0.08838834764831845f; // 1/sqrt(128)

    const float4 qv = ((const float4*)(q_rot + (size_t)(b * H_ + h) * D_))[lane];
    const float* Kb = k_cache + (size_t)(b * KVH_ + kvh) * S_CACHE_ * D_;
    const float* Vb = v_cache + (size_t)(b * KVH_ + kvh) * S_CACHE_ * D_;
    const float* kn = k_new + (size_t)(b * KVH_ + kvh) * D_;
    const float* vn = v_new + (size_t)(b * KVH_ + kvh) * D_;
    const int* msk = mask + (size_t)bh * NBLK_;

    float m = -INFINITY, l = 0.f;
    float4 acc = {0.f, 0.f, 0.f, 0.f};
    for (int blk = 0; blk < NBLK_; ++blk) {
        if (!msk[blk]) continue;
        #pragma unroll
        for (int j = 0; j < 2; ++j) {
            const int pos = blk * BLKSZ_ + wave * 2 + j;
            const float4* kr; const float4* vr;
            if (pos < S_CACHE_) {
                kr = (const float4*)(Kb + (size_t)pos * D_);
                vr = (const float4*)(Vb + (size_t)pos * D_);
            } else {            // position S-1 is the freshly projected token
                kr = (const float4*)kn;
                vr = (const float4*)vn;
            }
            const float4 kv = kr[lane];
            float s = qv.x * kv.x + qv.y * kv.y + qv.z * kv.z + qv.w * kv.w;
            for (int off = 16; off > 0; off >>= 1) s += __shfl_xor(s, off, 32);
            s *= scale;
            const float mn = fmaxf(m, s);
            const float corr = __expf(m - mn);
            const float p = __expf(s - mn);
            const float4 vv = vr[lane];
            acc.x = acc.x * corr + p * vv.x;
            acc.y = acc.y * corr + p * vv.y;
            acc.z = acc.z * corr + p * vv.z;
            acc.w = acc.w * corr + p * vv.w;
            l = l * corr + p;
            m = mn;
        }
    }
    __shared__ float sm[8], sl[8], sacc[8][128];
    if (lane == 0) { sm[wave] = m; sl[wave] = l; }
    ((float4*)sacc[wave])[lane] = acc;
    __syncthreads();
    if (tid < 128) {            // one thread per output dim
        float M = -INFINITY;
        #pragma unroll
        for (int w = 0; w < 8; ++w) M = fmaxf(M, sm[w]);
        float L = 0.f, A = 0.f;
        #pragma unroll
        for (int w = 0; w < 8; ++w) {
            const float c = __expf(sm[w] - M);
            L += sl[w] * c;
            A += sacc[w][tid] * c;
        }
        attnout[(size_t)(b * H_ + h) * D_ + tid] = A / L;
    }
}

// =====================================================================
// 6) output projection, split-K partials + deterministic combine
// =====================================================================
__global__ __launch_bounds__(128)
void oproj_partial_kernel(const float* __restrict__ attnout,
                          const float* __restrict__ Wo,
                          float* __restrict__ opart)   // [8][4][4096]
{
    __shared__ float ao[B_][128];
    const int g = blockIdx.x, hz = blockIdx.y, tid = threadIdx.x;
    const int col = g * 128 + tid;
    float acc0 = 0.f, acc1 = 0.f, acc2 = 0.f, acc3 = 0.f;
    const int hbeg = hz * 512;
    for (int h0 = hbeg; h0 < hbeg + 512; h0 += 128) {
        __syncthreads();
        for (int b = 0; b < B_; ++b) ao[b][tid] = attnout[b * HID_ + h0 + tid];
        __syncthreads();
        __builtin_prefetch(&Wo[(size_t)(h0 + 128) * HID_ + col], 0, 1);
        #pragma unroll 4
        for (int hh = 0; hh < 128; ++hh) {
            const float w = Wo[(size_t)(h0 + hh) * HID_ + col];
            acc0 += ao[0][hh] * w;
            acc1 += ao[1][hh] * w;
            acc2 += ao[2][hh] * w;
            acc3 += ao[3][hh] * w;
        }
    }
    opart[((size_t)hz * B_ + 0) * HID_ + col] = acc0;
    opart[((size_t)hz * B_ + 1) * HID_ + col] = acc1;
    opart[((size_t)hz * B_ + 2) * HID_ + col] = acc2;
    opart[((size_t)hz * B_ + 3) * HID_ + col] = acc3;
}

__global__ __launch_bounds__(128)
void oproj_combine_kernel(const float* __restrict__ opart, float* __restrict__ out)
{
    const int col = blockIdx.x * 128 + threadIdx.x;
    for (int b = 0; b < B_; ++b) {
        float s = 0.f;
        #pragma unroll
        for (int hz = 0; hz < 8; ++hz)
            s += opart[((size_t)hz * B_ + b) * HID_ + col];
        out[b * HID_ + col] = s;
    }
}

// =====================================================================
extern "C" void kernel_launch(void* const* d_in, const int* in_sizes, int n_in,
                              void* d_out, int out_size, void* d_ws, size_t ws_size,
                              hipStream_t stream)
{
    const float* hidden = (const float*)d_in[0];
    const float* k_cache = (const float*)d_in[1];
    const float* v_cache = (const float*)d_in[2];
    const float* attn_history = (const float*)d_in[3];
    const float* cosv = (const float*)d_in[4];
    const float* sinv = (const float*)d_in[5];
    const float* Wq = (const float*)d_in[6];
    const float* Wk = (const float*)d_in[7];
    const float* Wv = (const float*)d_in[8];
    const float* Wo = (const float*)d_in[9];
    const float* c1w = (const float*)d_in[10];
    const float* c1b = (const float*)d_in[11];
    const float* c2w = (const float*)d_in[12];
    const float* c2b = (const float*)d_in[13];
    const float* c3w = (const float*)d_in[14];
    const float* c3b = (const float*)d_in[15];

    float* ws = (float*)d_ws;
    float* qkv_part = ws;                  // 8*4*6144 = 196608 (reused by oproj)
    float* q_rot    = ws + 196608;         // 16384
    float* k_new    = ws + 212992;         // 4096
    float* v_new    = ws + 217088;         // 4096
    float* attnout  = ws + 221184;         // 16384
    float* tsp      = ws + 237568;         // 65536
    int*   selmask  = (int*)(ws + 303104); // 65536 ints  (total ~1.41 MB)

    qkv_partial_kernel<<<dim3(48, 8), 128, 0, stream>>>(hidden, Wq, Wk, Wv, qkv_part);
    rope_combine_kernel<<<48, 128, 0, stream>>>(qkv_part, cosv, sinv, q_rot, k_new, v_new);
    cnn_tsp_kernel<<<dim3(NBLK_ / TW, B_ * H_), 256, 0, stream>>>(
        attn_history, c1w, c1b, c2w, c2b, c3w, c3b, tsp);
    topk_mask_kernel<<<B_ * H_, 32, 0, stream>>>(tsp, selmask);
    sparse_attn_kernel<<<B_ * H_, 256, 0, stream>>>(
        k_cache, v_cache, q_rot, k_new, v_new, selmask, attnout);
    oproj_partial_kernel<<<dim3(32, 8), 128, 0, stream>>>(attnout, Wo, qkv_part);
    oproj_combine_kernel<<<32, 128, 0, stream>>>(qkv_part, (float*)d_out);
}